// DTFA_66675072303709
// MI455X (gfx1250) — compile-verified
//
#include <hip/hip_runtime.h>
#include <hip/hip_bf16.h>

// ---------------------------------------------------------------------------
// Problem constants (match reference)
// ---------------------------------------------------------------------------
#define BB  16
#define CC  64
#define OCC 16
#define TT  256
#define FF  256
#define NTHR 23

typedef float v2f __attribute__((ext_vector_type(2)));
typedef float v8f __attribute__((ext_vector_type(8)));

// threshold k (k = 1..23), folded at compile time to match
// float32(double(k*(k+1))/600.0) as the reference builds THRESHOLDS
#define THRK(i) ((float)((double)((i) * ((i) + 1)) / 600.0))

// ---------------------------------------------------------------------------
// Kernel 1: per-(b,c) tile means over T (-> mean_t[B,C,F]) and over F
// (-> mean_f[B,C,T]).  One block per (b,c); 256 threads.
// Pass 1 streams the tile coalesced (column sums); pass 2 re-reads the
// just-streamed 256 KB tile (L2-resident) for row sums.
// ---------------------------------------------------------------------------
__global__ __launch_bounds__(256) void mean_kernel(
    const float* __restrict__ x, float* __restrict__ mean_t,
    float* __restrict__ mean_f) {
  const int bc = blockIdx.x;
  const float* tile = x + (size_t)bc * (TT * FF);
  const int tid = threadIdx.x;

  // column sum (reduce over t) -> mean over T
  float cs = 0.f;
  for (int t = 0; t < TT; ++t) cs += tile[t * FF + tid];
  mean_t[bc * FF + tid] = cs * (1.0f / (float)TT);

  // row sum (reduce over f) -> mean over F
  float rs = 0.f;
  const float* row = tile + tid * FF;
  for (int f = 0; f < FF; f += 4) {
    float4 v = *(const float4*)(row + f);
    rs += v.x + v.y + v.z + v.w;
  }
  mean_f[bc * TT + tid] = rs * (1.0f / (float)FF);
}

// ---------------------------------------------------------------------------
// Kernel 2: SE branch, all in f32 WMMA (16x16x4).
//   H = relu(W1[16x64] * Z[64x256] + b1)     (16 k-steps of K=4)
//   S = sigmoid(W2[64x16] * H[16x256] + b2)  (4 k-steps of K=4)
// grid = (branch=2, b=16); block = 256 threads = 8 waves; each wave owns
// two 16-column tiles of L=256.  H tile bounced through per-wave LDS to
// convert the C/D layout into the B-operand layout.
// f32 fragment layouts (ISA 7.12.2): lanes 0-15 hold K(/M) low half,
// lanes 16-31 hold the high half; C/D vgpr r -> row r (lanes<16) / r+8.
// ---------------------------------------------------------------------------
__global__ __launch_bounds__(256) void se_kernel(
    const float* __restrict__ means,  // mean_t then mean_f, contiguous
    const float* __restrict__ w1, const float* __restrict__ b1,
    const float* __restrict__ w2, const float* __restrict__ b2,
    float* __restrict__ zbuf) {       // zt then zf, contiguous
  const int branch = blockIdx.x;      // 0: mean_t -> zt ; 1: mean_f -> zf
  const int b = blockIdx.y;
  const float* z = means + (size_t)branch * (BB * CC * 256) + (size_t)b * CC * 256;
  float* outp    = zbuf  + (size_t)branch * (BB * CC * 256) + (size_t)b * CC * 256;

  const unsigned tid  = threadIdx.x;
  const unsigned lane = tid & 31u;
  const unsigned w    = tid >> 5;       // wave id 0..7
  const unsigned hi   = lane >> 4;      // 0: lanes 0-15, 1: lanes 16-31
  const unsigned l15  = lane & 15u;

  __shared__ float Hl[8][16][16];       // per-wave 16x16 H tile

  for (int it = 0; it < 2; ++it) {
    const int nt  = (int)w + it * 8;    // N-tile 0..15
    const int col = nt * 16 + (int)l15;

    // --- GEMM1: H = W1 * Z ---
    v8f acc = {};
#pragma unroll
    for (int kk = 0; kk < 16; ++kk) {
      const int c0 = kk * 4 + (int)hi * 2;
      v2f a, bb;
      a.x = w1[l15 * CC + c0];
      a.y = w1[l15 * CC + c0 + 1];
      bb.x = z[c0 * 256 + col];
      bb.y = z[(c0 + 1) * 256 + col];
      acc = __builtin_amdgcn_wmma_f32_16x16x4_f32(false, a, false, bb,
                                                  (short)0, acc, false, false);
    }
    // bias + relu, park in LDS (row-major)
#pragma unroll
    for (int r = 0; r < 8; ++r) {
      const int m = r + 8 * (int)hi;
      Hl[w][m][l15] = fmaxf(acc[r] + b1[m], 0.0f);
    }
    __syncthreads();

    // --- GEMM2: S = W2 * H ---
#pragma unroll
    for (int mt = 0; mt < 4; ++mt) {
      v8f acc2 = {};
#pragma unroll
      for (int kk = 0; kk < 4; ++kk) {
        const int o0 = kk * 4 + (int)hi * 2;
        v2f a, bb;
        a.x = w2[(mt * 16 + (int)l15) * OCC + o0];
        a.y = w2[(mt * 16 + (int)l15) * OCC + o0 + 1];
        bb.x = Hl[w][o0][l15];
        bb.y = Hl[w][o0 + 1][l15];
        acc2 = __builtin_amdgcn_wmma_f32_16x16x4_f32(false, a, false, bb,
                                                     (short)0, acc2, false, false);
      }
#pragma unroll
      for (int r = 0; r < 8; ++r) {
        const int crow = mt * 16 + r + 8 * (int)hi;
        float s = acc2[r] + b2[crow];
        s = 1.0f / (1.0f + expf(-s));
        outp[crow * 256 + col] = s;
      }
    }
    __syncthreads();
  }
}

// ---------------------------------------------------------------------------
// Kernel 3: prefix sums of wf2 over the threshold axis:
//   P[c][m] = sum_{j<m} wf2[c][j],  m = 0..23   (24 entries per channel)
// ---------------------------------------------------------------------------
__global__ __launch_bounds__(64) void pfx_kernel(const float* __restrict__ wf2,
                                                 float* __restrict__ pfx) {
  const int c = threadIdx.x;
  float acc = 0.f;
  pfx[c * 24 + 0] = 0.f;
  for (int m = 1; m <= NTHR; ++m) {
    acc += wf2[c * NTHR + (m - 1)];
    pfx[c * 24 + m] = acc;
  }
}

// ---------------------------------------------------------------------------
// Kernel 4: gate  g[b,t,f] = sum_c wf[c]*zf[b,c,t]*zt[b,c,f] + bf
// as a 256x256x64 f32 WMMA GEMM per batch: G = (diag(wf)*Zf)^T * Zt.
// Also counts m = #{k : g > thr_k} ONCE per (b,t,f) and stores it as a byte,
// so the 67M-element final pass never touches the 23 thresholds.
// grid = (t-tile=16, b=16); 8 waves x 2 f-tiles each.
// ---------------------------------------------------------------------------
__global__ __launch_bounds__(256) void g_kernel(
    const float* __restrict__ zt, const float* __restrict__ zf,
    const float* __restrict__ wf, const float* __restrict__ bf,
    float* __restrict__ gbuf, unsigned char* __restrict__ mbuf) {
  const int mt = blockIdx.x;   // t-tile
  const int b  = blockIdx.y;
  const unsigned tid  = threadIdx.x;
  const unsigned lane = tid & 31u;
  const unsigned w    = tid >> 5;
  const unsigned hi   = lane >> 4;
  const unsigned l15  = lane & 15u;

  const float* zfb = zf + (size_t)b * CC * TT;
  const float* ztb = zt + (size_t)b * CC * FF;
  const int trow = mt * 16 + (int)l15;

  // A fragments (M=t, K=c), pre-scaled by wf[c]; reused across all f-tiles
  v2f afr[16];
#pragma unroll
  for (int kk = 0; kk < 16; ++kk) {
    const int c0 = kk * 4 + (int)hi * 2;
    afr[kk].x = wf[c0]     * zfb[c0 * TT + trow];
    afr[kk].y = wf[c0 + 1] * zfb[(c0 + 1) * TT + trow];
  }
  const float bfv = bf[0];

  for (int it = 0; it < 2; ++it) {
    const int ft   = (int)w + it * 8;
    const int fcol = ft * 16 + (int)l15;
    v8f acc = {};
#pragma unroll
    for (int kk = 0; kk < 16; ++kk) {
      const int c0 = kk * 4 + (int)hi * 2;
      v2f bb;
      bb.x = ztb[c0 * FF + fcol];
      bb.y = ztb[(c0 + 1) * FF + fcol];
      acc = __builtin_amdgcn_wmma_f32_16x16x4_f32(false, afr[kk], false, bb,
                                                  (short)0, acc, false, false);
    }
#pragma unroll
    for (int r = 0; r < 8; ++r) {
      const int t = mt * 16 + r + 8 * (int)hi;
      const float gv = acc[r] + bfv;
      int cnt = 0;
#pragma unroll
      for (int k = 1; k <= NTHR; ++k) cnt += (gv > THRK(k)) ? 1 : 0;
      const size_t idx = ((size_t)b * TT + t) * FF + fcol;
      gbuf[idx] = gv;
      mbuf[idx] = (unsigned char)cnt;
    }
  }
}

// ---------------------------------------------------------------------------
// Kernel 5: the bandwidth pass (reads 268MB x + 4MB g + 1MB m, writes 268MB):
//   out[b,c,t,f] = (g[b,t,f] * P[c][m[b,t,f]] + bf2[c]) * x[b,c,t,f]
// One block per (b,c); P[c][0..23] lives in LDS; float4 streaming.
// ---------------------------------------------------------------------------
__global__ __launch_bounds__(256) void final_kernel(
    const float* __restrict__ x, const float* __restrict__ gbuf,
    const unsigned char* __restrict__ mbuf, const float* __restrict__ pfx,
    const float* __restrict__ bf2, float* __restrict__ outp) {
  const int bc = blockIdx.x;
  const int b  = bc >> 6;      // /CC
  const int c  = bc & 63;      // %CC

  __shared__ float P[24];
  if (threadIdx.x < 24) P[threadIdx.x] = pfx[c * 24 + threadIdx.x];
  __syncthreads();

  const float bias  = bf2[c];
  const size_t xbase = (size_t)bc * (TT * FF);
  const size_t gbase = (size_t)b  * (TT * FF);

  for (int i = 0; i < (TT * FF) / (256 * 4); ++i) {
    const int e = i * 1024 + (int)threadIdx.x * 4;
    const float4 xv = *(const float4*)(x + xbase + e);
    const float4 gv = *(const float4*)(gbuf + gbase + e);
    const uchar4 mv = *(const uchar4*)(mbuf + gbase + e);
    float4 o;
    o.x = (gv.x * P[mv.x] + bias) * xv.x;
    o.y = (gv.y * P[mv.y] + bias) * xv.y;
    o.z = (gv.z * P[mv.z] + bias) * xv.z;
    o.w = (gv.w * P[mv.w] + bias) * xv.w;
    *(float4*)(outp + xbase + e) = o;
  }
}

// ---------------------------------------------------------------------------
// Launch
// ---------------------------------------------------------------------------
extern "C" void kernel_launch(void* const* d_in, const int* in_sizes, int n_in,
                              void* d_out, int out_size, void* d_ws, size_t ws_size,
                              hipStream_t stream) {
  const float* x   = (const float*)d_in[0];
  const float* w1  = (const float*)d_in[1];
  const float* b1  = (const float*)d_in[2];
  const float* w2  = (const float*)d_in[3];
  const float* b2  = (const float*)d_in[4];
  const float* wf  = (const float*)d_in[5];
  const float* bf  = (const float*)d_in[6];
  const float* wf2 = (const float*)d_in[7];
  const float* bf2 = (const float*)d_in[8];
  float* out = (float*)d_out;

  float* ws = (float*)d_ws;
  float* mean_t = ws;                         // B*C*F = 262144 floats
  float* mean_f = ws + 262144;                // B*C*T = 262144 (contiguous!)
  float* zbuf   = ws + 524288;                // zt then zf: 2*262144
  float* gbuf   = ws + 1048576;               // B*T*F = 1048576
  float* pfx    = ws + 2097152;               // 64*24 = 1536 (pad to 2048)
  unsigned char* mbuf = (unsigned char*)(ws + 2097152 + 2048);  // B*T*F bytes

  mean_kernel<<<BB * CC, 256, 0, stream>>>(x, mean_t, mean_f);
  se_kernel<<<dim3(2, BB), 256, 0, stream>>>(mean_t, w1, b1, w2, b2, zbuf);
  pfx_kernel<<<1, 64, 0, stream>>>(wf2, pfx);
  g_kernel<<<dim3(TT / 16, BB), 256, 0, stream>>>(zbuf, zbuf + 262144, wf, bf,
                                                  gbuf, mbuf);
  final_kernel<<<BB * CC, 256, 0, stream>>>(x, gbuf, mbuf, pfx, bf2, out);
}